// DynamicOptimizerModule_4767413698605
// MI455X (gfx1250) — compile-verified
//
#include <hip/hip_runtime.h>

// Elementwise DAG collapses to:
//   out[0..N)   = loss + prev_loss + p0 + p1 + p2 + p3 + p4 + p5
//   out[N..2N)  = same value (slot15 = slot14)
//
// Memory-bound: 512 MB read + 128 MB write = 640 MB -> ~27.5 us floor @ 23.3 TB/s.
// Strategy: single pass, float4 (b128) non-temporal loads/stores, sum in regs,
// dual store (never re-read the sum). No WMMA: there is no matrix contraction.

typedef float v4f __attribute__((ext_vector_type(4)));

__global__ __launch_bounds__(256) void sum8_dual_store_kernel(
    const float* __restrict__ a0, const float* __restrict__ a1,
    const float* __restrict__ a2, const float* __restrict__ a3,
    const float* __restrict__ a4, const float* __restrict__ a5,
    const float* __restrict__ a6, const float* __restrict__ a7,
    float* __restrict__ out, long long n)   // out has 2*n elements
{
    const long long n4 = n >> 2;  // number of float4 elements
    const long long i4 = (long long)blockIdx.x * blockDim.x + threadIdx.x;

    if (i4 < n4) {
        const long long off = i4 << 2;
        v4f s;
        s  = __builtin_nontemporal_load((const v4f*)(a0 + off));
        s += __builtin_nontemporal_load((const v4f*)(a1 + off));
        s += __builtin_nontemporal_load((const v4f*)(a2 + off));
        s += __builtin_nontemporal_load((const v4f*)(a3 + off));
        s += __builtin_nontemporal_load((const v4f*)(a4 + off));
        s += __builtin_nontemporal_load((const v4f*)(a5 + off));
        s += __builtin_nontemporal_load((const v4f*)(a6 + off));
        s += __builtin_nontemporal_load((const v4f*)(a7 + off));
        // slot14
        __builtin_nontemporal_store(s, (v4f*)(out + off));
        // slot15 (identical)
        __builtin_nontemporal_store(s, (v4f*)(out + n + off));
    }

    // Scalar tail for n % 4 != 0 (not hit for the 16*1024*1024 shape, but keeps
    // the kernel correct for any size). Only the very first thread does this.
    if (blockIdx.x == 0 && threadIdx.x == 0) {
        for (long long j = n4 << 2; j < n; ++j) {
            float t = a0[j] + a1[j] + a2[j] + a3[j] +
                      a4[j] + a5[j] + a6[j] + a7[j];
            out[j]     = t;
            out[n + j] = t;
        }
    }
}

extern "C" void kernel_launch(void* const* d_in, const int* in_sizes, int n_in,
                              void* d_out, int out_size, void* d_ws, size_t ws_size,
                              hipStream_t stream) {
    (void)n_in; (void)d_ws; (void)ws_size; (void)out_size;

    const float* a0 = (const float*)d_in[0];  // loss
    const float* a1 = (const float*)d_in[1];  // prev_loss
    const float* a2 = (const float*)d_in[2];  // p0
    const float* a3 = (const float*)d_in[3];  // p1
    const float* a4 = (const float*)d_in[4];  // p2
    const float* a5 = (const float*)d_in[5];  // p3
    const float* a6 = (const float*)d_in[6];  // p4
    const float* a7 = (const float*)d_in[7];  // p5
    float* out = (float*)d_out;

    const long long n  = (long long)in_sizes[0];  // 16*1024*1024
    const long long n4 = n >> 2;

    const int  block = 256;                         // 8 wave32s
    const long long grid64 = (n4 + block - 1) / block;
    const unsigned grid = (unsigned)grid64;

    sum8_dual_store_kernel<<<grid, block, 0, stream>>>(
        a0, a1, a2, a3, a4, a5, a6, a7, out, n);
}